// EpisodicMemory_36180804501648
// MI455X (gfx1250) — compile-verified
//
#include <hip/hip_runtime.h>
#include <math.h>

// EpisodicMemory on MI455X (gfx1250, wave32).
// Bandwidth-bound: values (1.07GB) + keys (0.27GB) streamed once -> ~29-58us floor.
// GEMMs use V_WMMA_F32_16X16X4_F32 to stay in fp32 (reference dtype).

typedef float v2f __attribute__((ext_vector_type(2)));
typedef float v8f __attribute__((ext_vector_type(8)));

#define B_   512
#define S_   1024
#define KDIM 128
#define VDIM 512
#define INV_SCALE 0.08838834764831845f   // 1/sqrt(128)

// ---------------------------------------------------------------------------
// GEMM: C[m,n] = epilogue( sum_k A[m,k] * W[n,k] + bias[n] )
// One wave (32 threads) per 16x16 output tile. fp32 WMMA 16x16x4.
// A-matrix VGPR layout (ISA 7.12.2, 32-bit 16x4): m = lane%16;
//   lanes 0-15 hold {K=0,K=1}, lanes 16-31 hold {K=2,K=3}.
// => lane loads exactly its b64 pair at k + 2*(lane>=16): no cndmask needed.
// B-matrix (4x16) mirrors with n = lane%16. C/D: vgpr r -> M = r + 8*(lane>=16).
// Two accumulators (even/odd K-chunk) halve the WMMA dependency chain.
// ---------------------------------------------------------------------------
enum Epi { EPI_BIAS = 0, EPI_SILU = 1, EPI_GATE = 2 };

template <int EPI>
__global__ __launch_bounds__(32) void wmma_gemm_xwt(
    const float* __restrict__ A, int lda,
    const float* __restrict__ W, int ldw,
    const float* __restrict__ bias,
    float* __restrict__ C, int ldc,
    int Kd,
    const float* __restrict__ hid,    // EPI_GATE only
    const float* __restrict__ hcat)   // EPI_GATE only (retrieved at +VDIM, ld 2V)
{
  const int lane  = threadIdx.x & 31;
  const int l16   = lane & 15;
  const int hi    = lane >> 4;                  // 0: K0/K1 half, 1: K2/K3 half
  const int mBase = blockIdx.y * 16;
  const int nBase = blockIdx.x * 16;

  const float* __restrict__ arow = A + (size_t)(mBase + l16) * lda + 2 * hi;
  const float* __restrict__ wrow = W + (size_t)(nBase + l16) * ldw + 2 * hi;

  v8f acc0 = {};
  v8f acc1 = {};
  for (int k = 0; k < Kd; k += 8) {          // Kd is a multiple of 8
    const v2f a0 = *(const v2f*)(arow + k);
    const v2f b0 = *(const v2f*)(wrow + k);
    const v2f a1 = *(const v2f*)(arow + k + 4);
    const v2f b1 = *(const v2f*)(wrow + k + 4);
    acc0 = __builtin_amdgcn_wmma_f32_16x16x4_f32(false, a0, false, b0,
                                                 (short)0, acc0, false, false);
    acc1 = __builtin_amdgcn_wmma_f32_16x16x4_f32(false, a1, false, b1,
                                                 (short)0, acc1, false, false);
  }
  const v8f acc = acc0 + acc1;

  const float bn = bias[nBase + l16];
#pragma unroll
  for (int r = 0; r < 8; ++r) {
    const int m = mBase + r + hi * 8;
    const int n = nBase + l16;
    float x = acc[r] + bn;
    if (EPI == EPI_SILU) {
      x = x / (1.0f + __expf(-x));                       // silu
    } else if (EPI == EPI_GATE) {
      const float g  = 1.0f / (1.0f + __expf(-x));       // sigmoid gate
      const float rt = hcat[(size_t)m * (2 * VDIM) + VDIM + n];
      x = hid[(size_t)m * VDIM + n] + g * rt;            // hidden + gate*retrieved
    }
    C[(size_t)m * ldc + n] = x;
  }
}

// ---------------------------------------------------------------------------
// Attention: one block (256 threads / 8 waves) per batch row.
// Phase 1: each wave dots one key row per step (coalesced 512B row reads),
//          ring-buffer substitution when s == write_ptr, skip rows >= filled+1.
// Phase 2: block softmax in LDS.
// Phase 3: stream values rows with b128 loads: two 128-thread parity groups
//          each own alternating rows (4KB = 2 rows per iteration, coalesced),
//          substitution applied as a correction in the owning group, partials
//          combined through LDS. Writes h_cat = [hidden | retrieved].
// ---------------------------------------------------------------------------
__global__ __launch_bounds__(256) void attn_kernel(
    const float* __restrict__ keys, const float* __restrict__ values,
    const float* __restrict__ keyv, const float* __restrict__ valv,
    const float* __restrict__ hidden,
    const int* __restrict__ write_ptr, const int* __restrict__ filled,
    const float* __restrict__ query,
    float* __restrict__ hcat)
{
  __shared__ __align__(16) float sc[S_];
  __shared__ __align__(16) float comb[VDIM];
  __shared__ float red[8];

  const int b    = blockIdx.x;
  const int tid  = threadIdx.x;
  const int lane = tid & 31;
  const int wave = tid >> 5;

  const int wp = write_ptr[b];
  int fl = filled[b] + 1;
  if (fl > S_) fl = S_;

  // per-lane query chunk: q[b, lane*4 .. lane*4+3]
  const float4 qv = *(const float4*)(query + (size_t)b * KDIM + lane * 4);

  // ---- scores ----
  for (int s = wave; s < S_; s += 8) {
    if (s < fl) {
      const float* krow = (s == wp) ? (keyv + (size_t)b * KDIM)
                                    : (keys + ((size_t)b * S_ + s) * KDIM);
      const float4 kv = *(const float4*)(krow + lane * 4);
      float p = kv.x * qv.x + kv.y * qv.y + kv.z * qv.z + kv.w * qv.w;
#pragma unroll
      for (int off = 16; off > 0; off >>= 1) p += __shfl_xor(p, off, 32);
      if (lane == 0) sc[s] = p * INV_SCALE;
    } else {
      if (lane == 0) sc[s] = -INFINITY;
    }
  }
  __syncthreads();

  // ---- block max ----
  float mx = -INFINITY;
  for (int s = tid; s < S_; s += 256) mx = fmaxf(mx, sc[s]);
#pragma unroll
  for (int off = 16; off > 0; off >>= 1) mx = fmaxf(mx, __shfl_xor(mx, off, 32));
  if (lane == 0) red[wave] = mx;
  __syncthreads();
  mx = red[0];
#pragma unroll
  for (int i = 1; i < 8; ++i) mx = fmaxf(mx, red[i]);

  // ---- exp + sum (sc left UNNORMALIZED; 1/sum applied at the end) ----
  float psum = 0.0f;
  for (int s = tid; s < S_; s += 256) {
    const float e = __expf(sc[s] - mx);   // -inf rows -> 0
    sc[s] = e;
    psum += e;
  }
  __syncthreads();                         // red reads done, sc writes published
#pragma unroll
  for (int off = 16; off > 0; off >>= 1) psum += __shfl_xor(psum, off, 32);
  if (lane == 0) red[wave] = psum;
  __syncthreads();
  float tot = red[0];
#pragma unroll
  for (int i = 1; i < 8; ++i) tot += red[i];
  const float inv = (tot > 0.0f) ? (1.0f / tot) : 0.0f;   // nan_to_num behavior

  // ---- retrieved: b128 streaming, two parity groups over rows ----
  const int cg = (tid & 127) * 4;          // column base: 128 threads x 4 = V
  const int p  = tid >> 7;                 // row parity group: 0 or 1
  const float* __restrict__ vb = values + (size_t)b * S_ * VDIM + cg;
  float4 acc = {0.0f, 0.0f, 0.0f, 0.0f};
#pragma unroll 4
  for (int s = p; s < fl; s += 2) {
    const float  w = sc[s];
    const float4 x = *(const float4*)(vb + (size_t)s * VDIM);
    acc.x += w * x.x;
    acc.y += w * x.y;
    acc.z += w * x.z;
    acc.w += w * x.w;
  }
  if (wp < fl && ((wp & 1) == p)) {        // ring-buffer substitution correction
    const float  w   = sc[wp];
    const float4 old = *(const float4*)(vb + (size_t)wp * VDIM);
    const float4 nw  = *(const float4*)(valv + (size_t)b * VDIM + cg);
    acc.x += w * (nw.x - old.x);
    acc.y += w * (nw.y - old.y);
    acc.z += w * (nw.z - old.z);
    acc.w += w * (nw.w - old.w);
  }
  if (p == 1) *(float4*)(comb + cg) = acc;
  __syncthreads();

  float* __restrict__ hc = hcat + (size_t)b * (2 * VDIM);
  if (p == 0) {
    const float4 o = *(const float4*)(comb + cg);
    float4 r;
    r.x = (acc.x + o.x) * inv;
    r.y = (acc.y + o.y) * inv;
    r.z = (acc.z + o.z) * inv;
    r.w = (acc.w + o.w) * inv;
    *(float4*)(hc + VDIM + cg) = r;        // h_cat[:, V:2V] = retrieved
  }

  // ---- h_cat[:, 0:V] = hidden (all 256 threads, float2 each) ----
  const int v0 = tid * 2;
  const float2 h = *(const float2*)(hidden + (size_t)b * VDIM + v0);
  *(float2*)(hc + v0) = h;
}

// ---------------------------------------------------------------------------
extern "C" void kernel_launch(void* const* d_in, const int* in_sizes, int n_in,
                              void* d_out, int out_size, void* d_ws, size_t ws_size,
                              hipStream_t stream) {
  const float* keys      = (const float*)d_in[0];
  const float* values    = (const float*)d_in[1];
  const float* key       = (const float*)d_in[2];
  const float* value     = (const float*)d_in[3];
  const float* hidden    = (const float*)d_in[4];
  const int*   write_ptr = (const int*)d_in[5];
  const int*   filled    = (const int*)d_in[6];
  const float* Wq        = (const float*)d_in[7];
  const float* bq        = (const float*)d_in[8];
  const float* Wg1       = (const float*)d_in[9];
  const float* bg1       = (const float*)d_in[10];
  const float* Wg2       = (const float*)d_in[11];
  const float* bg2       = (const float*)d_in[12];
  const float* Wo        = (const float*)d_in[13];
  const float* bo        = (const float*)d_in[14];
  float* out = (float*)d_out;

  float* ws   = (float*)d_ws;
  float* q    = ws;                            // B*K  = 65536 floats
  float* hcat = q    + (size_t)B_ * KDIM;      // B*2V = 524288
  float* g    = hcat + (size_t)B_ * 2 * VDIM;  // B*V  = 262144
  float* pre  = g    + (size_t)B_ * VDIM;      // B*V  = 262144

  // 1) query = hidden @ Wq.T + bq           (M=512, N=128, K=512)
  wmma_gemm_xwt<EPI_BIAS><<<dim3(KDIM / 16, B_ / 16), 32, 0, stream>>>(
      hidden, VDIM, Wq, VDIM, bq, q, KDIM, VDIM, nullptr, nullptr);

  // 2) attention (ring-buffer write folded in) -> h_cat
  attn_kernel<<<B_, 256, 0, stream>>>(keys, values, key, value, hidden,
                                      write_ptr, filled, q, hcat);

  // 3) g = silu(h_cat @ Wg1.T + bg1)        (M=512, N=512, K=1024)
  wmma_gemm_xwt<EPI_SILU><<<dim3(VDIM / 16, B_ / 16), 32, 0, stream>>>(
      hcat, 2 * VDIM, Wg1, 2 * VDIM, bg1, g, VDIM, 2 * VDIM, nullptr, nullptr);

  // 4) pre = hidden + sigmoid(g @ Wg2.T + bg2) * retrieved   (M=512,N=512,K=512)
  wmma_gemm_xwt<EPI_GATE><<<dim3(VDIM / 16, B_ / 16), 32, 0, stream>>>(
      g, VDIM, Wg2, VDIM, bg2, pre, VDIM, VDIM, hidden, hcat);

  // 5) out = pre @ Wo.T + bo                (M=512, N=512, K=512)
  wmma_gemm_xwt<EPI_BIAS><<<dim3(VDIM / 16, B_ / 16), 32, 0, stream>>>(
      pre, VDIM, Wo, VDIM, bo, out, VDIM, VDIM, nullptr, nullptr);
}